// BottleneckFFN_63101659513386
// MI455X (gfx1250) — compile-verified
//
#include <hip/hip_runtime.h>
#include <math.h>

// ---------------------------------------------------------------------------
// BottleneckFFN for MI455X (gfx1250, wave32):
//   h = x @ W1^T + b1 ; h = gelu_exact(h) ; g = h @ W2^T + b2 ; y = LN(g)*gamma+beta
// x: [NTOK, 256] fp32, OUT = 8.  HBM-bound (~4 FLOP/byte, ~268 MB in).
// Staging: Tensor Data Mover (tensor_load_to_lds) streams 16x256-f32 tiles
// into LDS with hardware row padding (1040B stride -> conflict-free ds reads).
// Reduction: v_wmma_f32_16x16x4_f32, 64 per tile, b1 folded into C init.
// ---------------------------------------------------------------------------

typedef float v2f __attribute__((ext_vector_type(2)));
typedef float v4f __attribute__((ext_vector_type(4)));
typedef float v8f __attribute__((ext_vector_type(8)));
typedef unsigned int u32x4 __attribute__((ext_vector_type(4)));
typedef int i32x4 __attribute__((ext_vector_type(4)));
typedef int i32x8 __attribute__((ext_vector_type(8)));

#define DIM 256
#define OUT 8
#define WAVES 2
#define TPB (WAVES * 32)
#define KPAIRS (DIM / 2)           // 128 k-pairs
#define A_ROW_BYTES 1040           // 256 floats + 16B TDM pad -> bank = t*4 (mod 64)
#define A_ROW_PAIRS (A_ROW_BYTES / 8)   // 130
#define H_STRIDE 20                // epilogue row stride (floats), conflict-free

__global__ __launch_bounds__(TPB) void ffn_wmma_kernel(
    const float* __restrict__ x,
    const float* __restrict__ W1,
    const float* __restrict__ b1,
    const float* __restrict__ W2,
    const float* __restrict__ b2,
    const float* __restrict__ gamma,
    const float* __restrict__ beta,
    float* __restrict__ out,
    int ntok)
{
    // B (W1^T zero-padded to 16 cols), k-pair layout: Bp[kpair][n] = {W1[n][2k], W1[n][2k+1]}
    __shared__ v2f ldsB[KPAIRS * 16];
    // Per-wave A tile: 16 tokens x 256 floats, 1040B row stride (TDM-padded)
    __shared__ __align__(16) unsigned char ldsA[WAVES][16 * A_ROW_BYTES];

    const int tid  = threadIdx.x;
    const int lane = tid & 31;
    const int wave = tid >> 5;
    const int n    = lane & 15;    // fragment column / token id
    const int hi   = lane >> 4;    // K sub-pair selector

    // ---- preamble: W1 -> LDS pair layout (cols 8..15 zero) ----
    for (int idx = tid; idx < KPAIRS * 16; idx += TPB) {
        const int kp = idx >> 4;
        const int nn = idx & 15;
        v2f v = {0.0f, 0.0f};
        if (nn < OUT) {
            v.x = W1[nn * DIM + 2 * kp + 0];
            v.y = W1[nn * DIM + 2 * kp + 1];
        }
        ldsB[idx] = v;
    }
    __syncthreads();

    const int wv   = __builtin_amdgcn_readfirstlane(wave);
    const int tile = blockIdx.x * WAVES + wv;
    if (tile * 16 + 15 >= ntok) return;     // full tiles only (NTOK % 16 == 0)

    const int tok0 = tile * 16;
    const float* __restrict__ xt = x + (size_t)tok0 * DIM;

#if __has_builtin(__builtin_amdgcn_tensor_load_to_lds)
    {
        // ---- TDM: DMA 16x256-f32 tile, pad 4 DWORDs after every 256 DWORDs ----
        const unsigned long long ga = (unsigned long long)(uintptr_t)xt;
        unsigned lds_off = (unsigned)(unsigned long long)(uintptr_t)&ldsA[wv][0];
        lds_off = __builtin_amdgcn_readfirstlane(lds_off);

        u32x4 g0;
        g0.x = 1u;                                    // count=1, user descriptor
        g0.y = lds_off;                               // lds_addr (bytes)
        g0.z = (unsigned)ga;                          // global_addr[31:0]
        g0.w = (unsigned)((ga >> 32) & 0x01FFFFFFu)   // global_addr[56:32]
             | (2u << 30);                            // type = 2 ("image")
        i32x8 g1;
        g1[0] = (2 << 16)      // data_size = 4 bytes
              | (1 << 20)      // pad_enable
              | (7 << 22)      // pad_interval: 256 DWORDs
              | (3 << 25);     // pad_amount:   4 DWORDs (16B)
        g1[1] = (DIM & 0xFFFF) << 16;                 // tensor_dim0 = 256 (lo16)
        g1[2] = (DIM >> 16) | (16 << 16);             // tensor_dim0 hi / tensor_dim1 = 16
        g1[3] = (DIM << 16);                          // tile_dim0 = 256
        g1[4] = 16;                                   // tile_dim1 = 16, tile_dim2 = 0
        g1[5] = DIM;                                  // tensor_dim0_stride = 256
        g1[6] = 0;
        g1[7] = 0;
        i32x4 gz4 = {0, 0, 0, 0};
        i32x8 gz8 = {0, 0, 0, 0, 0, 0, 0, 0};

        // amdgpu-toolchain (clang-23) 6-arg form
        __builtin_amdgcn_tensor_load_to_lds(g0, g1, gz4, gz4, gz8, 0);
        __builtin_amdgcn_s_wait_tensorcnt(0);
        asm volatile("" ::: "memory");                // keep ds reads below the wait
    }
#else
    {
        // Fallback: manual coalesced staging (512B per instruction per row)
        v4f* Aw = (v4f*)&ldsA[wave][0];
        for (int t = 0; t < 16; ++t) {
            v4f v = *(const v4f*)(xt + (size_t)t * DIM + lane * 4);
            Aw[t * (A_ROW_BYTES / 16) + lane] = v;
        }
        v4f* Aw2 = (v4f*)&ldsA[wave][0];
        for (int t = 0; t < 16; ++t) {
            v4f v = *(const v4f*)(xt + (size_t)t * DIM + 128 + lane * 4);
            Aw2[t * (A_ROW_BYTES / 16) + 32 + lane] = v;
        }
    }
#endif

    // C init: fold b1 (broadcast down each column; padded cols -> 0)
    const float cb = (n < OUT) ? b1[n] : 0.0f;
    v8f c = {cb, cb, cb, cb, cb, cb, cb, cb};

    const v2f* __restrict__ Aw = (const v2f*)&ldsA[wave][0];

    // ---- K reduction: 64 x v_wmma_f32_16x16x4_f32 ----
    #pragma unroll
    for (int kk = 0; kk < DIM; kk += 4) {
        const int kp = (kk >> 1) + hi;
        v2f a = Aw[n * A_ROW_PAIRS + kp];        // A: lane(t=n,hi) = x[t][kk+2hi..+1]
        v2f b = ldsB[kp * 16 + n];               // B: lane(n,hi)   = W1[n][kk+2hi..+1]
        c = __builtin_amdgcn_wmma_f32_16x16x4_f32(
                false, a, false, b, (short)0, c, false, false);
    }

    // ---- transpose D fragment through LDS (stride 20 -> conflict-free) ----
    float* hrow = (float*)&ldsA[wave][0];        // reuse private tile region
    #pragma unroll
    for (int v = 0; v < 8; ++v) {
        const int m = v + hi * 8;                // token row in this VGPR
        hrow[m * H_STRIDE + n] = c[v];
    }

    // ---- per-token epilogue on lanes 0..15 ----
    if (hi == 0) {
        const int t = n;
        float h[OUT];
        #pragma unroll
        for (int o = 0; o < OUT; ++o) {
            const float v = hrow[t * H_STRIDE + o];
            h[o] = 0.5f * v * (1.0f + erff(v * 0.70710678118654752440f)); // exact GELU
        }

        float g[OUT];
        float mu = 0.0f;
        #pragma unroll
        for (int p = 0; p < OUT; ++p) {
            float acc = b2[p];
            #pragma unroll
            for (int o = 0; o < OUT; ++o)
                acc = fmaf(W2[p * OUT + o], h[o], acc);
            g[p] = acc;
            mu += acc;
        }
        mu *= (1.0f / OUT);

        float var = 0.0f;
        #pragma unroll
        for (int p = 0; p < OUT; ++p) {
            const float d = g[p] - mu;
            var = fmaf(d, d, var);
        }
        var *= (1.0f / OUT);
        const float r = rsqrtf(var + 1e-5f);

        v4f o0, o1;
        #pragma unroll
        for (int p = 0; p < 4; ++p) {
            o0[p] = fmaf((g[p]     - mu) * r, gamma[p],     beta[p]);
            o1[p] = fmaf((g[p + 4] - mu) * r, gamma[p + 4], beta[p + 4]);
        }
        float* op = out + (size_t)(tok0 + t) * OUT;
        *(v4f*)(op)     = o0;    // 16 lanes x 32B contiguous -> coalesced
        *(v4f*)(op + 4) = o1;
    }
}

extern "C" void kernel_launch(void* const* d_in, const int* in_sizes, int n_in,
                              void* d_out, int out_size, void* d_ws, size_t ws_size,
                              hipStream_t stream) {
    const float* x     = (const float*)d_in[0];
    const float* W1    = (const float*)d_in[1];
    const float* b1    = (const float*)d_in[2];
    const float* W2    = (const float*)d_in[3];
    const float* b2    = (const float*)d_in[4];
    const float* gamma = (const float*)d_in[5];
    const float* beta  = (const float*)d_in[6];
    float* out = (float*)d_out;

    const int ntok  = in_sizes[0] / DIM;         // 128*2048 = 262144
    const int tiles = ntok / 16;                 // 16384 wave-tiles
    const int blocks = (tiles + WAVES - 1) / WAVES;

    ffn_wmma_kernel<<<blocks, TPB, 0, stream>>>(x, W1, b1, W2, b2, gamma, beta,
                                                out, ntok);
}